// PatialChamferDistanceL1_58342835749039
// MI455X (gfx1250) — compile-verified
//
#include <hip/hip_runtime.h>
#include <math.h>

typedef __attribute__((ext_vector_type(2))) float v2f;
typedef __attribute__((ext_vector_type(8))) float v8f;

#define B_    4
#define N_    8192
#define M_    8192
#define MT    (M_ / 16)             // 512 column tiles per batch
#define WAVES 8
#define NITER (MT / WAVES)          // 64 column tiles per wave
#define BLOCKS (B_ * N_ / 32)       // 1024 blocks, 32 xyz1 rows each

// Pack xyz1 -> A rows (-2x,-2y,-2z,1) + squared norms; xyz2 -> B cols (x,y,z,||p||^2).
__global__ void prep_kernel(const float* __restrict__ xyz1,
                            const float* __restrict__ xyz2,
                            float* __restrict__ a4,
                            float* __restrict__ n1,
                            float* __restrict__ b4) {
    int i = blockIdx.x * blockDim.x + threadIdx.x;
    if (i >= B_ * N_) return;   // B*N == B*M here
    {
        float x = xyz1[3 * i + 0], y = xyz1[3 * i + 1], z = xyz1[3 * i + 2];
        a4[4 * i + 0] = -2.0f * x;
        a4[4 * i + 1] = -2.0f * y;
        a4[4 * i + 2] = -2.0f * z;
        a4[4 * i + 3] = 1.0f;
        n1[i] = x * x + y * y + z * z;
    }
    {
        float x = xyz2[3 * i + 0], y = xyz2[3 * i + 1], z = xyz2[3 * i + 2];
        b4[4 * i + 0] = x;
        b4[4 * i + 1] = y;
        b4[4 * i + 2] = z;
        b4[4 * i + 3] = x * x + y * y + z * z;
    }
}

__global__ void init_kernel(float* acc) {
    if (threadIdx.x < 2) acc[threadIdx.x] = 0.0f;
}

// One block per 32-row tile of xyz1 (two 16-row WMMA tiles sharing each B fragment);
// 8 waves split the 512 column tiles of xyz2.
__global__ void __launch_bounds__(WAVES * 32)
chamfer_kernel(const float* __restrict__ a4,
               const float* __restrict__ n1,
               const float* __restrict__ b4,
               const float* __restrict__ w,
               float* __restrict__ acc) {
    __shared__ float smin[WAVES][32];

    const int tid  = threadIdx.x;
    const int lane = tid & 31;
    const int wave = tid >> 5;
    const int half = lane >> 4;   // K-pair / row-half selector (per ISA fragment layouts)
    const int n    = lane & 15;   // point index within a 16-tile

    const int batch = blockIdx.x / (N_ / 32);
    const int r0    = blockIdx.x * 32;        // global xyz1 row base (32 rows)

    // Two A fragments (rows r0..r0+15 and r0+16..r0+31).
    v2f afrag0 = *(const v2f*)(a4 + (size_t)(r0 + n) * 4 + half * 2);
    v2f afrag1 = *(const v2f*)(a4 + (size_t)(r0 + 16 + n) * 4 + half * 2);

    // Per-wave B base: column tile index = wave + it*WAVES; 1 tile = 64 floats.
    const float* bb = b4 + (size_t)batch * M_ * 4
                         + (size_t)(wave * 16 + n) * 4 + half * 2;

    float acc0[8], acc1[8];
#pragma unroll
    for (int r = 0; r < 8; ++r) { acc0[r] = 3.0e38f; acc1[r] = 3.0e38f; }

    // Constant trip count -> scalar loop, no exec masking around the WMMAs.
    for (int it = 0; it < NITER; it += 2) {
        v2f bfa = *(const v2f*)(bb + (size_t)(it + 0) * WAVES * 64);
        v2f bfb = *(const v2f*)(bb + (size_t)(it + 1) * WAVES * 64);

        v8f z = {};
        v8f c0a = __builtin_amdgcn_wmma_f32_16x16x4_f32(false, afrag0, false, bfa,
                                                        (short)0, z, false, false);
        v8f c1a = __builtin_amdgcn_wmma_f32_16x16x4_f32(false, afrag1, false, bfa,
                                                        (short)0, z, false, false);
        v8f c0b = __builtin_amdgcn_wmma_f32_16x16x4_f32(false, afrag0, false, bfb,
                                                        (short)0, z, false, false);
        v8f c1b = __builtin_amdgcn_wmma_f32_16x16x4_f32(false, afrag1, false, bfb,
                                                        (short)0, z, false, false);

#pragma unroll
        for (int r = 0; r < 8; ++r) {
            asm("v_min3_num_f32 %0, %1, %2, %3"
                : "=v"(acc0[r]) : "0"(acc0[r]), "v"(c0a[r]), "v"(c0b[r]));
            asm("v_min3_num_f32 %0, %1, %2, %3"
                : "=v"(acc1[r]) : "0"(acc1[r]), "v"(c1a[r]), "v"(c1b[r]));
        }
    }

    // Min over the 16 columns held by each 16-lane half.
#pragma unroll
    for (int m = 1; m < 16; m <<= 1) {
#pragma unroll
        for (int r = 0; r < 8; ++r) {
            acc0[r] = fminf(acc0[r], __shfl_xor(acc0[r], m, 16));
            acc1[r] = fminf(acc1[r], __shfl_xor(acc1[r], m, 16));
        }
    }
    if (n == 0) {
#pragma unroll
        for (int r = 0; r < 8; ++r) {
            smin[wave][half * 8 + r]      = acc0[r];   // rows r0      + (r + 8*half)
            smin[wave][16 + half * 8 + r] = acc1[r];   // rows r0 + 16 + (r + 8*half)
        }
    }
    __syncthreads();

    // Wave 0: reduce across waves, finish per-row math, accumulate block sums.
    if (tid < 32) {
        float m = smin[0][tid];
#pragma unroll
        for (int wv = 1; wv < WAVES; ++wv) m = fminf(m, smin[wv][tid]);
        float d2  = m + n1[r0 + tid];            // + ||p1||^2
        float d   = sqrtf(fmaxf(d2, 0.0f));
        float wt  = w[r0 + tid];
        float num = d * wt;
#pragma unroll
        for (int s = 1; s < 32; s <<= 1) {
            num += __shfl_xor(num, s, 32);
            wt  += __shfl_xor(wt, s, 32);
        }
        if (tid == 0) {
            atomicAdd(&acc[0], num);   // sum of weighted distances
            atomicAdd(&acc[1], wt);    // sum of weights
        }
    }
}

__global__ void final_kernel(const float* acc, float* out) {
    if (threadIdx.x == 0) out[0] = acc[0] / acc[1];
}

extern "C" void kernel_launch(void* const* d_in, const int* in_sizes, int n_in,
                              void* d_out, int out_size, void* d_ws, size_t ws_size,
                              hipStream_t stream) {
    const float* xyz1 = (const float*)d_in[0];  // [B,N,3]
    const float* xyz2 = (const float*)d_in[1];  // [B,M,3]
    const float* w    = (const float*)d_in[2];  // [B,N]
    float* out = (float*)d_out;

    // Workspace layout (floats): acc[2] (padded to 256B) | n1[B*N] | a4[4*B*N] | b4[4*B*M]
    float* acc = (float*)d_ws;
    float* n1  = acc + 64;
    float* a4  = n1 + (size_t)B_ * N_;
    float* b4  = a4 + (size_t)4 * B_ * N_;

    init_kernel<<<1, 32, 0, stream>>>(acc);
    prep_kernel<<<(B_ * N_ + 255) / 256, 256, 0, stream>>>(xyz1, xyz2, a4, n1, b4);
    chamfer_kernel<<<BLOCKS, WAVES * 32, 0, stream>>>(a4, n1, b4, w, acc);
    final_kernel<<<1, 32, 0, stream>>>(acc, out);
}